// Soft_PQ_46402826666370
// MI455X (gfx1250) — compile-verified
//
#include <hip/hip_runtime.h>
#include <hip/hip_bf16.h>

typedef __attribute__((ext_vector_type(2))) float v2f;
typedef __attribute__((ext_vector_type(8))) float v8f;

#define PQ_N      16384
#define PQ_W      64          // codewords
#define PQ_B      128         // books
#define PQ_L      32          // dims per book
#define PQ_D      (PQ_B * PQ_L)   // 4096
#define PQ_TAU    10.0f
#define CB_PITCH  34          // padded LDS pitch for cb tile (bank-conflict avoidance)
#define SF_PITCH  66          // padded LDS pitch for soft tile

// ---------------------------------------------------------------------------
// Pass 1: per-(word, book) L2-normalize codewords; store normalized C and
// post-normalization ||c||^2 (matches reference: c2 computed AFTER clamp-div).
// ---------------------------------------------------------------------------
__global__ __launch_bounds__(256) void pq_norm_kernel(
    const float* __restrict__ C, float* __restrict__ Cn, float* __restrict__ c2) {
  int t = blockIdx.x * 256 + threadIdx.x;   // 8192 threads total
  int w = t & (PQ_W - 1);
  int b = t >> 6;
  const float4* src = (const float4*)(C + (size_t)w * PQ_D + b * PQ_L);
  float4*       dst = (float4*)(Cn + (size_t)w * PQ_D + b * PQ_L);
  float4 q[8];
  float ss = 0.f;
#pragma unroll
  for (int i = 0; i < 8; ++i) {
    q[i] = src[i];
    ss += q[i].x * q[i].x + q[i].y * q[i].y + q[i].z * q[i].z + q[i].w * q[i].w;
  }
  float inv = 1.f / fmaxf(sqrtf(ss), 1e-12f);
  float s2 = 0.f;
#pragma unroll
  for (int i = 0; i < 8; ++i) {
    float4 n;
    n.x = q[i].x * inv; n.y = q[i].y * inv; n.z = q[i].z * inv; n.w = q[i].w * inv;
    s2 += n.x * n.x + n.y * n.y + n.z * n.z + n.w * n.w;
    dst[i] = n;
  }
  c2[b * PQ_W + w] = s2;
}

// ---------------------------------------------------------------------------
// Pass 2: one block = (book b, 64 rows). 4 waves, each wave owns 16 rows.
//   GEMM1 (WMMA f32 16x16x4): xc[16x64] = x_tile[16x32] * cb^T[32x64]
//   softmax + argmin over 64 words (in D-layout, half-wave shuffles)
//   GEMM2 (WMMA f32 16x16x4): Z[16x32] = soft[16x64] * cb[64x32]
// ---------------------------------------------------------------------------
__global__ __launch_bounds__(128) void pq_main_kernel(
    const float* __restrict__ x, const float* __restrict__ Cn,
    const float* __restrict__ c2, float* __restrict__ Z,
    float* __restrict__ outIdx) {
  __shared__ float cb[PQ_W * CB_PITCH];          // 64 words x 32 dims (padded)
  __shared__ float softs[4 * 16 * SF_PITCH];     // per-wave 16x64 soft tile

  const int tid  = threadIdx.x;
  const int wid  = tid >> 5;
  const int lane = tid & 31;
  const int m16  = lane & 15;      // column-within-tile / row-id for A loads
  const int h    = lane >> 4;      // half-wave select (K-pair select)

  const int b  = blockIdx.x & (PQ_B - 1);
  const int n0 = (blockIdx.x >> 7) * 64;

  // ---- cooperative load of normalized codebook tile into LDS ----
  const float* cbsrc = Cn + (size_t)b * PQ_L;
#pragma unroll
  for (int i = 0; i < 4; ++i) {
    int q  = tid + 128 * i;        // 512 float4 slots
    int w  = q >> 3;
    int c4 = (q & 7) * 4;
    float4 v = *(const float4*)(cbsrc + (size_t)w * PQ_D + c4);
    cb[w * CB_PITCH + c4 + 0] = v.x;
    cb[w * CB_PITCH + c4 + 1] = v.y;
    cb[w * CB_PITCH + c4 + 2] = v.z;
    cb[w * CB_PITCH + c4 + 3] = v.w;
  }
  __syncthreads();

  // per-lane ||c||^2 for the word this lane holds in each accumulator group
  float c2v[4];
#pragma unroll
  for (int g = 0; g < 4; ++g) c2v[g] = c2[b * PQ_W + g * 16 + m16];

  // ---- load x tile in WMMA A-layout (v0 = K+2h, v1 = K+2h+1 per 4-step) ----
  const int   grow = n0 + wid * 16 + m16;
  const float* xrow = x + (size_t)grow * PQ_D + b * PQ_L;
  v2f a[8];
  float x2p = 0.f;
#pragma unroll
  for (int s = 0; s < 8; ++s) {
    a[s] = *(const v2f*)(xrow + 4 * s + 2 * h);
    x2p += a[s].x * a[s].x + a[s].y * a[s].y;
  }
  float x2full = x2p + __shfl_xor(x2p, 16, 32);  // full ||x_row||^2, row = m16

  // ---- GEMM1: xc = x * cb^T ----
  v8f acc[4];
#pragma unroll
  for (int g = 0; g < 4; ++g) acc[g] = (v8f){0.f,0.f,0.f,0.f,0.f,0.f,0.f,0.f};
#pragma unroll
  for (int g = 0; g < 4; ++g) {
    const float* cbw = &cb[(g * 16 + m16) * CB_PITCH];
#pragma unroll
    for (int s = 0; s < 8; ++s) {
      v2f bv = *(const v2f*)(cbw + 4 * s + 2 * h);   // B[k, word] = cb[word, k]
      acc[g] = __builtin_amdgcn_wmma_f32_16x16x4_f32(
          false, a[s], false, bv, (short)0, acc[g], false, false);
    }
  }

  // broadcast ||x||^2 of output rows (r + 8h) to all lanes of each half
  float x2r[8];
#pragma unroll
  for (int r = 0; r < 8; ++r) x2r[r] = __shfl(x2full, r + 8 * h, 32);

  // scores s = -tau*dist = 2*tau*xc - tau*(||x||^2 + ||c||^2)
#pragma unroll
  for (int g = 0; g < 4; ++g)
#pragma unroll
    for (int r = 0; r < 8; ++r)
      acc[g][r] = 2.f * PQ_TAU * acc[g][r] - PQ_TAU * (x2r[r] + c2v[g]);

  // ---- per-row softmax over 64 words + argmin(dist) = argmax(score) ----
  float* sl = &softs[wid * 16 * SF_PITCH];
#pragma unroll
  for (int r = 0; r < 8; ++r) {
    float vm = acc[0][r];
    int   im = m16;
#pragma unroll
    for (int g = 1; g < 4; ++g) {
      float v = acc[g][r];
      int   i = g * 16 + m16;
      if (v > vm) { vm = v; im = i; }
    }
#pragma unroll
    for (int off = 1; off < 16; off <<= 1) {       // stays within half-wave
      float ov = __shfl_xor(vm, off, 32);
      int   oi = __shfl_xor(im, off, 32);
      if (ov > vm || (ov == vm && oi < im)) { vm = ov; im = oi; }
    }
    // vm is also the row max for a stable softmax
    float se = 0.f;
#pragma unroll
    for (int g = 0; g < 4; ++g) {
      acc[g][r] = __expf(acc[g][r] - vm);
      se += acc[g][r];
    }
#pragma unroll
    for (int off = 1; off < 16; off <<= 1) se += __shfl_xor(se, off, 32);
    float rinv = 1.f / se;
#pragma unroll
    for (int g = 0; g < 4; ++g) acc[g][r] *= rinv;
    if (m16 == 0)
      outIdx[(size_t)(n0 + wid * 16 + r + 8 * h) * PQ_B + b] = (float)im;
  }

  // ---- transpose soft through LDS: D-layout -> A-layout ----
#pragma unroll
  for (int g = 0; g < 4; ++g)
#pragma unroll
    for (int r = 0; r < 8; ++r)
      sl[(r + 8 * h) * SF_PITCH + g * 16 + m16] = acc[g][r];
  __syncthreads();   // wave-private data, but make ordering explicit

  // ---- GEMM2: Z[16x32] = soft[16x64] * cb[64x32] ----
  v8f z0 = (v8f){0.f,0.f,0.f,0.f,0.f,0.f,0.f,0.f};
  v8f z1 = (v8f){0.f,0.f,0.f,0.f,0.f,0.f,0.f,0.f};
  const float* srow = &sl[m16 * SF_PITCH];
#pragma unroll
  for (int s = 0; s < 16; ++s) {
    v2f av = *(const v2f*)(srow + 4 * s + 2 * h);   // A[m, k]: words along K
    int w0 = 4 * s + 2 * h;
    v2f b0v, b1v;
    b0v.x = cb[(w0    ) * CB_PITCH + m16];
    b0v.y = cb[(w0 + 1) * CB_PITCH + m16];
    b1v.x = cb[(w0    ) * CB_PITCH + 16 + m16];
    b1v.y = cb[(w0 + 1) * CB_PITCH + 16 + m16];
    z0 = __builtin_amdgcn_wmma_f32_16x16x4_f32(false, av, false, b0v,
                                               (short)0, z0, false, false);
    z1 = __builtin_amdgcn_wmma_f32_16x16x4_f32(false, av, false, b1v,
                                               (short)0, z1, false, false);
  }

  // ---- store Z (coalesced across the 16 lanes of each half) ----
#pragma unroll
  for (int r = 0; r < 8; ++r) {
    size_t row = (size_t)(n0 + wid * 16 + r + 8 * h);
    float* zp = Z + row * PQ_D + b * PQ_L;
    zp[m16]      = z0[r];
    zp[16 + m16] = z1[r];
  }
}

// ---------------------------------------------------------------------------
extern "C" void kernel_launch(void* const* d_in, const int* in_sizes, int n_in,
                              void* d_out, int out_size, void* d_ws, size_t ws_size,
                              hipStream_t stream) {
  const float* x = (const float*)d_in[0];   // [16384, 4096]
  const float* C = (const float*)d_in[1];   // [64, 4096]

  float* Cn = (float*)d_ws;                 // [64, 4096] normalized codebook
  float* c2 = Cn + (size_t)PQ_W * PQ_D;     // [128, 64]  post-norm ||c||^2

  float* Z    = (float*)d_out;                        // [16384, 4096]
  float* oidx = Z + (size_t)PQ_N * PQ_D;              // [16384, 128] (as float)

  pq_norm_kernel<<<dim3((PQ_W * PQ_B) / 256), dim3(256), 0, stream>>>(C, Cn, c2);
  pq_main_kernel<<<dim3((PQ_N / 64) * PQ_B), dim3(128), 0, stream>>>(x, Cn, c2, Z, oidx);
}